// CPCUnsupersivedCriterion_23218593202458
// MI455X (gfx1250) — compile-verified
//
#include <hip/hip_runtime.h>
#include <math.h>

typedef __attribute__((ext_vector_type(2))) float v2f;
typedef __attribute__((ext_vector_type(8))) float v8f;

#define NP    12      // N_PREDICTS
#define DIMAR 256
#define DIME  256
#define NNEG  128
#define NGT   8
#define SEQ   128
#define WIN   116     // SEQ - NP
#define MTOT  (NGT*WIN)  // 928

// ---------------- zero the atomic accumulators ----------------
__global__ void k_zero(float* __restrict__ acc) {
    int i = blockIdx.x * blockDim.x + threadIdx.x;
    if (i < 192) acc[i] = 0.0f;
}

// ---------------- locC[k][m][e] = sum_d c[m][d] * W[k][e][d] ----------------
// grid = (58 mtiles, 4, 12 k), block = 128 (4 waves, one e-tile each)
__global__ void k_locc(const float* __restrict__ cFeat,
                       const float* __restrict__ W,
                       float* __restrict__ locC) {
    const int lane  = threadIdx.x & 31;
    const int wave  = threadIdx.x >> 5;
    const int mtile = blockIdx.x;              // 0..57  (58*16 == 928)
    const int etile = blockIdx.y * 4 + wave;   // 0..15
    const int k     = blockIdx.z;              // 0..11

    // A tile 16x4 f32: lanes 0-15 hold K={0,1}, lanes 16-31 hold K={2,3}
    const int mrow = mtile * 16 + (lane & 15);
    const int b    = mrow / WIN;
    const int t    = mrow % WIN;
    const int kb   = (lane >> 4) * 2;
    const float* aptr = cFeat + (size_t)(b * SEQ + t) * DIMAR + kb;

    // B tile 4x16: B[d][e] = W[k][e][d]  (contiguous in d -> float2 loads)
    const int ecol = etile * 16 + (lane & 15);
    const float* bptr = W + (size_t)(k * DIME + ecol) * DIMAR + kb;

    v8f acc = {};
    #pragma unroll 4
    for (int d0 = 0; d0 < DIMAR; d0 += 4) {
        v2f a  = *(const v2f*)(aptr + d0);
        v2f bb = *(const v2f*)(bptr + d0);
        acc = __builtin_amdgcn_wmma_f32_16x16x4_f32(
            false, a, false, bb, (short)0, acc, false, false);
    }

    // C layout: VGPR j -> M = j + 8*(lane>=16), N = lane&15
    const int mbase = mtile * 16 + 8 * (lane >> 4);
    const int e     = etile * 16 + (lane & 15);
    float* out = locC + (size_t)k * MTOT * DIME + e;
    #pragma unroll
    for (int j = 0; j < 8; ++j)
        out[(size_t)(mbase + j) * DIME] = acc[j];
}

// ---------------- per-(b,t): scores, softmax, loss/acc accumulation ----------------
// grid = 928 blocks (bt), block = 128 threads (4 waves x 2 n-tiles of 16)
__global__ void k_score(const float* __restrict__ locC,
                        const float* __restrict__ gt,
                        const float* __restrict__ pool,
                        const long long* __restrict__ extIdx,
                        float* __restrict__ lossAcc,
                        float* __restrict__ corrAcc) {
    const int bt   = blockIdx.x;          // b*WIN + t
    const int b    = bt / WIN;
    const int t    = bt % WIN;
    const int tid  = threadIdx.x;
    const int lane = tid & 31;
    const int wave = tid >> 5;

    __shared__ float S[NP][NNEG];   // neg scores (already /256)
    __shared__ float red[NP][8];    // partial reductions
    __shared__ float P[NP];         // pos scores (/256)
    __shared__ float MN[NP];        // max over negs
    __shared__ float MA[NP];        // overall max

    // A: locC rows k=0..11 (padded to 16 with a zero mask; keep EXEC all-1s)
    const int kk = lane & 15;
    const int eb = (lane >> 4) * 2;
    const int ksafe = (kk < NP) ? kk : 0;
    const float zmask = (kk < NP) ? 1.0f : 0.0f;
    const float* aptr = locC + (size_t)(ksafe * MTOT + bt) * DIME + eb;

    // B: two gathered negative rows per lane (index hoisted out of K loop)
    const int n0 = (wave * 2 + 0) * 16 + (lane & 15);
    const int n1 = (wave * 2 + 1) * 16 + (lane & 15);
    const long long i0 = extIdx[(size_t)(b * NNEG + n0) * WIN + t];
    const long long i1 = extIdx[(size_t)(b * NNEG + n1) * WIN + t];
    const float* b0 = pool + (size_t)i0 * DIME + eb;
    const float* b1 = pool + (size_t)i1 * DIME + eb;
    __builtin_prefetch(b0, 0, 0);
    __builtin_prefetch(b1, 0, 0);

    v8f acc0 = {}, acc1 = {};
    #pragma unroll 4
    for (int e0 = 0; e0 < DIME; e0 += 4) {
        v2f a = *(const v2f*)(aptr + e0);
        a.x *= zmask; a.y *= zmask;
        v2f x0 = *(const v2f*)(b0 + e0);
        v2f x1 = *(const v2f*)(b1 + e0);
        acc0 = __builtin_amdgcn_wmma_f32_16x16x4_f32(
            false, a, false, x0, (short)0, acc0, false, false);
        acc1 = __builtin_amdgcn_wmma_f32_16x16x4_f32(
            false, a, false, x1, (short)0, acc1, false, false);
    }

    // spill C to LDS: VGPR j -> k = j + 8*(lane>=16), n = tile*16 + (lane&15)
    const int krow = 8 * (lane >> 4);
    #pragma unroll
    for (int j = 0; j < 8; ++j) {
        int k = krow + j;
        if (k < NP) {
            S[k][n0] = acc0[j] * (1.0f / 256.0f);
            S[k][n1] = acc1[j] * (1.0f / 256.0f);
        }
    }
    __syncthreads();

    // pos dot: pos[k] = <locC[k,b,t,:], gt[b, t+k+1, :]> / 256
    if (tid < 96) {
        int k = tid >> 3, s = tid & 7;
        const float* lc = locC + (size_t)(k * MTOT + bt) * DIME;
        const float* gp = gt + (size_t)(b * SEQ + t + k + 1) * DIME;
        float sum = 0.0f;
        #pragma unroll 8
        for (int e = s * 32; e < s * 32 + 32; ++e) sum += lc[e] * gp[e];
        red[k][s] = sum;
    }
    __syncthreads();
    if (tid < NP) {
        float sum = 0.0f;
        for (int s = 0; s < 8; ++s) sum += red[tid][s];
        P[tid] = sum * (1.0f / 256.0f);
    }
    __syncthreads();

    // max over negatives
    if (tid < 96) {
        int k = tid >> 3, s = tid & 7;
        float m = -INFINITY;
        for (int n = s * 16; n < s * 16 + 16; ++n) m = fmaxf(m, S[k][n]);
        red[k][s] = m;
    }
    __syncthreads();
    if (tid < NP) {
        float m = red[tid][0];
        for (int s = 1; s < 8; ++s) m = fmaxf(m, red[tid][s]);
        MN[tid] = m;
        MA[tid] = fmaxf(m, P[tid]);
    }
    __syncthreads();

    // sum of exp
    if (tid < 96) {
        int k = tid >> 3, s = tid & 7;
        float m = MA[k], sum = 0.0f;
        for (int n = s * 16; n < s * 16 + 16; ++n) sum += __expf(S[k][n] - m);
        red[k][s] = sum;
    }
    __syncthreads();
    if (tid < NP) {
        int k = tid;
        float sum = __expf(P[k] - MA[k]);
        for (int s = 0; s < 8; ++s) sum += red[k][s];
        float lse = MA[k] + __logf(sum);
        atomicAdd(&lossAcc[k * NGT + b], lse - P[k]);
        if (P[k] >= MN[k])                    // argmax == 0 (ties -> index 0)
            atomicAdd(&corrAcc[k * NGT + b], 1.0f);
    }
}

// ---------------- finalize: losses (mean over t), acc ( /(nGt*window) ) ----------------
__global__ void k_final(const float* __restrict__ lossAcc,
                        const float* __restrict__ corrAcc,
                        float* __restrict__ out) {
    int i = threadIdx.x;
    if (i < 96) {
        out[i]      = lossAcc[i] * (1.0f / (float)WIN);
        out[96 + i] = corrAcc[i] * (1.0f / (float)(NGT * WIN));
    }
}

extern "C" void kernel_launch(void* const* d_in, const int* in_sizes, int n_in,
                              void* d_out, int out_size, void* d_ws, size_t ws_size,
                              hipStream_t stream) {
    const float*     cFeat  = (const float*)d_in[0];      // (8,128,256)
    const float*     gt     = (const float*)d_in[1];      // (8,128,256)
    const float*     pool   = (const float*)d_in[2];      // (8192,256)
    const float*     W      = (const float*)d_in[3];      // (12,256,256)
    const long long* extIdx = (const long long*)d_in[4];  // (118784,) int64

    float* ws      = (float*)d_ws;
    float* lossAcc = ws;            // 96
    float* corrAcc = ws + 96;       // 96
    float* locC    = ws + 256;      // 12*928*256 floats (~11.4 MB)

    k_zero<<<1, 256, 0, stream>>>(ws);

    dim3 g1(58, 4, 12);             // 58 m-tiles, 16 e-tiles (4 waves/block), 12 k
    k_locc<<<g1, 128, 0, stream>>>(cFeat, W, locC);

    k_score<<<MTOT, 128, 0, stream>>>(locC, gt, pool, extIdx, lossAcc, corrAcc);

    k_final<<<1, 128, 0, stream>>>(lossAcc, corrAcc, (float*)d_out);
}